// NodeLevelGAT_67439576482330
// MI455X (gfx1250) — compile-verified
//
#include <hip/hip_runtime.h>
#include <hip/hip_bf16.h>

typedef __attribute__((ext_vector_type(2))) float v2f;
typedef __attribute__((ext_vector_type(8))) float v8f;
typedef __attribute__((ext_vector_type(4))) unsigned int v4u;
typedef __attribute__((ext_vector_type(8))) int v8i;
typedef __attribute__((ext_vector_type(4))) int v4i;

#define N_NODES 50000
#define D_IN    256
#define D_OUT   128
#define LRELU_SLOPE 0.01f

// ---------------------------------------------------------------------------
// K0: init out / denom / max-keys (must run every call: we accumulate atomically)
// ---------------------------------------------------------------------------
__global__ void gat_init(float* __restrict__ out, unsigned* __restrict__ mkey,
                         float* __restrict__ denom, int n_out, int n_nodes) {
    int stride = gridDim.x * blockDim.x;
    for (int i = blockIdx.x * blockDim.x + threadIdx.x; i < n_out; i += stride)
        out[i] = 0.0f;
    for (int i = blockIdx.x * blockDim.x + threadIdx.x; i < n_nodes; i += stride) {
        // key(-inf): bits(-inf)=0xFF800000, negative => ~bits = 0x007FFFFF
        mkey[i]  = 0x007FFFFFu;
        denom[i] = 0.0f;
    }
}

// ---------------------------------------------------------------------------
// K1: z = h @ W^T via V_WMMA_F32_16X16X4_F32.
//  - W (128x256 f32 = 128 KB) staged whole into LDS via the Tensor Data Mover
//    (one 1-D TDM descriptor; wave 0 issues, s_wait_tensorcnt + barrier).
//  - One wave computes a 16(row) x 128(col) slab of z: 8 v8f accumulators.
//  - K loop (step 4) double-buffers A (global b64) and the 8 B fragments
//    (LDS b64) in registers so loads overlap the 8 WMMAs per step.
// Fragment layouts per ISA 7.12.2:
//   A (16x4):  VGPR j, lanes 0-15: (M=lane, K=j); lanes16-31: (M=lane-16, K=j+2)
//   B (4x16):  VGPR j, lanes 0-15: (K=j, N=lane); lanes16-31: (K=j+2, N=lane-16)
//   C/D 16x16: VGPR r, lanes 0-15: (M=r, N=lane); lanes16-31: (M=r+8, N=lane-16)
// ---------------------------------------------------------------------------
__global__ void gat_gemm_wmma(const float* __restrict__ h,
                              const float* __restrict__ W,
                              float* __restrict__ z) {
    __shared__ float Wlds[D_OUT * D_IN];   // 128 KB (WGP has 320 KB)

    // ---- Stage W -> LDS ----------------------------------------------------
#if defined(__has_builtin) && __has_builtin(__builtin_amdgcn_tensor_load_to_lds)
    if (threadIdx.x < 32) {                // wave 0 issues the TDM op once
        const unsigned long long ga = (unsigned long long)(uintptr_t)W;
        const unsigned lds_addr = (unsigned)(uintptr_t)(void*)Wlds; // low 32b = LDS byte addr
        // D# group 0: count=1, lds_addr, global_addr[56:0], type=2 ("image")
        v4u g0 = { 1u,
                   lds_addr,
                   (unsigned)(ga & 0xFFFFFFFFu),
                   (unsigned)((ga >> 32) & 0x01FFFFFFu) | 0x80000000u };
        // D# group 1: data_size=4B (code 2); 1-D tensor/tile of 32768 dwords
        //  w0: data_size<<16       w1: tensor_dim0[15:0]<<16 (=0x8000)
        //  w2: tensor_dim1=1<<16   w3: tile_dim0[15:0]<<16   (=0x8000)
        //  w5: tensor_dim0_stride = 32768
        v8i g1 = { (int)0x00020000, (int)0x80000000u, (int)0x00010000,
                   (int)0x80000000u, 0, 32768, 0, 0 };
        v4i gz4 = { 0, 0, 0, 0 };          // groups 2/3 unused (<=2-D tensor)
        v8i gz8 = { 0, 0, 0, 0, 0, 0, 0, 0 };
        __builtin_amdgcn_tensor_load_to_lds(g0, g1, gz4, gz4, gz8, 0);
        __builtin_amdgcn_s_wait_tensorcnt(0);
    }
#else
    {   // fallback: cooperative copy (float4)
        const float4* __restrict__ src4 = (const float4*)W;
        float4* __restrict__ dst4 = (float4*)Wlds;
        for (int i = threadIdx.x; i < (D_OUT * D_IN) / 4; i += blockDim.x)
            dst4[i] = src4[i];
    }
#endif
    __syncthreads();

    const int wavesPerBlock = blockDim.x >> 5;
    const int wave = blockIdx.x * wavesPerBlock + (threadIdx.x >> 5);
    const int lane = threadIdx.x & 31;
    const int nMTiles = N_NODES / 16;            // 3125, exact
    if (wave >= nMTiles) return;                 // wave-uniform: EXEC stays all-ones

    const int m0    = wave * 16;
    const int half  = lane >> 4;                 // 0 or 1
    const int sub   = lane & 15;
    const int khalf = half * 2;                  // K offset for this lane half

    v8f acc[8];
#pragma unroll
    for (int nt = 0; nt < 8; ++nt) acc[nt] = (v8f){0.f,0.f,0.f,0.f,0.f,0.f,0.f,0.f};

    const float* __restrict__ hrow = h + (size_t)(m0 + sub) * D_IN + khalf;

    v2f Ab[2];
    v2f Bb[2][8];

    // prime buffer 0 (k0 = 0)
    {
        const float2 t = *(const float2*)(hrow);
        Ab[0].x = t.x; Ab[0].y = t.y;
#pragma unroll
        for (int nt = 0; nt < 8; ++nt) {
            const float2 b = *(const float2*)(&Wlds[(nt * 16 + sub) * D_IN + khalf]);
            Bb[0][nt].x = b.x; Bb[0][nt].y = b.y;
        }
    }

#pragma unroll 2
    for (int k0 = 0; k0 < D_IN; k0 += 4) {
        const int cur = (k0 >> 2) & 1;
        const int nxt = cur ^ 1;
        const int k1  = k0 + 4;
        if (k1 < D_IN) {
            const float2 t = *(const float2*)(hrow + k1);
            Ab[nxt].x = t.x; Ab[nxt].y = t.y;
#pragma unroll
            for (int nt = 0; nt < 8; ++nt) {
                const float2 b = *(const float2*)(&Wlds[(nt * 16 + sub) * D_IN + k1 + khalf]);
                Bb[nxt][nt].x = b.x; Bb[nxt][nt].y = b.y;
            }
        }
#pragma unroll
        for (int nt = 0; nt < 8; ++nt) {
            acc[nt] = __builtin_amdgcn_wmma_f32_16x16x4_f32(
                /*neg_a=*/false, Ab[cur], /*neg_b=*/false, Bb[cur][nt],
                /*c_mod=*/(short)0, acc[nt], /*reuse_a=*/false, /*reuse_b=*/false);
        }
    }

    // Store D tiles
#pragma unroll
    for (int nt = 0; nt < 8; ++nt) {
        const int col = nt * 16 + sub;
#pragma unroll
        for (int r = 0; r < 8; ++r) {
            const int row = m0 + r + half * 8;
            z[(size_t)row * D_OUT + col] = acc[nt][r];
        }
    }
}

// ---------------------------------------------------------------------------
// K2: es[n] = z[n] . a[0:128], ed[n] = z[n] . a[128:256].  One wave per node.
// ---------------------------------------------------------------------------
__global__ void gat_rowdots(const float* __restrict__ z, const float* __restrict__ a,
                            float* __restrict__ es, float* __restrict__ ed) {
    const int node = blockIdx.x * (blockDim.x >> 5) + (threadIdx.x >> 5);
    const int lane = threadIdx.x & 31;
    if (node >= N_NODES) return;

    const float4 zv = ((const float4*)(z + (size_t)node * D_OUT))[lane];
    const float4 as = ((const float4*)(a))[lane];
    const float4 ad = ((const float4*)(a + D_OUT))[lane];
    float s = zv.x*as.x + zv.y*as.y + zv.z*as.z + zv.w*as.w;
    float d = zv.x*ad.x + zv.y*ad.y + zv.z*ad.z + zv.w*ad.w;
#pragma unroll
    for (int off = 16; off > 0; off >>= 1) {
        s += __shfl_down(s, off, 32);
        d += __shfl_down(d, off, 32);
    }
    if (lane == 0) { es[node] = s; ed[node] = d; }
}

// ---------------------------------------------------------------------------
// K3: per edge e = leaky_relu(es[src]+ed[dst]); segment-max via uint-key atomicMax
// ---------------------------------------------------------------------------
__device__ __forceinline__ unsigned f32_to_key(float f) {
    unsigned u = __float_as_uint(f);
    return (u & 0x80000000u) ? ~u : (u | 0x80000000u);
}
__device__ __forceinline__ float key_to_f32(unsigned u) {
    return (u & 0x80000000u) ? __uint_as_float(u & 0x7FFFFFFFu) : __uint_as_float(~u);
}

__global__ void gat_edge_e(const int* __restrict__ src, const int* __restrict__ dst,
                           const float* __restrict__ es, const float* __restrict__ ed,
                           float* __restrict__ eedge, unsigned* __restrict__ mkey,
                           int nE) {
    const int i = blockIdx.x * blockDim.x + threadIdx.x;
    if (i >= nE) return;
    float v = es[src[i]] + ed[dst[i]];
    v = (v > 0.0f) ? v : LRELU_SLOPE * v;
    eedge[i] = v;
    atomicMax(mkey + dst[i], f32_to_key(v));
}

// ---------------------------------------------------------------------------
// K4: exp_e = exp(e - m[dst]); denom[dst] += exp_e
// ---------------------------------------------------------------------------
__global__ void gat_edge_exp(const int* __restrict__ dst, float* __restrict__ eedge,
                             const unsigned* __restrict__ mkey,
                             float* __restrict__ denom, int nE) {
    const int i = blockIdx.x * blockDim.x + threadIdx.x;
    if (i >= nE) return;
    const int d = dst[i];
    const float m  = key_to_f32(mkey[d]);
    const float ex = __expf(eedge[i] - m);
    eedge[i] = ex;
    atomicAdd(denom + d, ex);
}

// ---------------------------------------------------------------------------
// K5: out[dst] += (exp_e/denom[dst]) * z[src].  One wave per edge; lane handles
// 4 consecutive floats (float4 gather + 4x global_atomic_add_f32 scatter).
// ---------------------------------------------------------------------------
__global__ void gat_scatter(const int* __restrict__ src, const int* __restrict__ dst,
                            const float* __restrict__ eedge, const float* __restrict__ denom,
                            const float* __restrict__ z, float* __restrict__ out, int nE) {
    const int edge = blockIdx.x * (blockDim.x >> 5) + (threadIdx.x >> 5);
    const int lane = threadIdx.x & 31;
    if (edge >= nE) return;
    const int s = src[edge];
    const int d = dst[edge];
    const float alpha = eedge[edge] / denom[d];
    const float4 v = ((const float4*)(z + (size_t)s * D_OUT))[lane];
    float* o = out + (size_t)d * D_OUT + lane * 4;
    atomicAdd(o + 0, alpha * v.x);
    atomicAdd(o + 1, alpha * v.y);
    atomicAdd(o + 2, alpha * v.z);
    atomicAdd(o + 3, alpha * v.w);
}

// ---------------------------------------------------------------------------
extern "C" void kernel_launch(void* const* d_in, const int* in_sizes, int n_in,
                              void* d_out, int out_size, void* d_ws, size_t ws_size,
                              hipStream_t stream) {
    const float* h   = (const float*)d_in[0];    // 50000 x 256
    const float* W   = (const float*)d_in[1];    // 128 x 256
    const float* a   = (const float*)d_in[2];    // 256
    const int*   src = (const int*)d_in[3];      // 800000
    const int*   dst = (const int*)d_in[4];      // 800000
    const int nE = in_sizes[3];
    float* out = (float*)d_out;                  // 50000 x 128

    // Workspace carve-up
    float*    z     = (float*)d_ws;                        // 6,400,000 f
    float*    es    = z + (size_t)N_NODES * D_OUT;         // 50,000 f
    float*    ed    = es + N_NODES;                        // 50,000 f
    unsigned* mkey  = (unsigned*)(ed + N_NODES);           // 50,000 u32
    float*    denom = (float*)(mkey + N_NODES);            // 50,000 f
    float*    eedge = denom + N_NODES;                     // nE f

    // K0: init
    gat_init<<<2048, 256, 0, stream>>>(out, mkey, denom, N_NODES * D_OUT, N_NODES);

    // K1: z = h @ W^T  (3125 waves, 4 waves/block; W staged via TDM -> LDS)
    {
        const int nWaves = N_NODES / 16;                   // 3125
        const int wpb = 4;
        gat_gemm_wmma<<<(nWaves + wpb - 1) / wpb, wpb * 32, 0, stream>>>(h, W, z);
    }

    // K2: row dots (50000 waves, 8 waves/block)
    {
        const int wpb = 8;
        gat_rowdots<<<(N_NODES + wpb - 1) / wpb, wpb * 32, 0, stream>>>(z, a, es, ed);
    }

    // K3: edge scores + segment max
    gat_edge_e<<<(nE + 255) / 256, 256, 0, stream>>>(src, dst, es, ed, eedge, mkey, nE);

    // K4: exp + segment sum
    gat_edge_exp<<<(nE + 255) / 256, 256, 0, stream>>>(dst, eedge, mkey, denom, nE);

    // K5: weighted scatter-add (one wave per edge, 8 waves/block)
    {
        const int wpb = 8;
        gat_scatter<<<(nE + wpb - 1) / wpb, wpb * 32, 0, stream>>>(src, dst, eedge, denom, z, out, nE);
    }
}